// FFNO_12910671691928
// MI455X (gfx1250) — compile-verified
//
#include <hip/hip_runtime.h>
#include <math.h>

// ---------------------------------------------------------------------------
// FFNO forward for gfx1250 (MI455X).  fp32 throughout; matrix ops via
// V_WMMA_F32_16X16X4_F32.  All activations (33.5MB each) are L2-resident.
// ---------------------------------------------------------------------------

typedef __attribute__((ext_vector_type(2))) float v2f;
typedef __attribute__((ext_vector_type(8))) float v8f;

#define GRIDN 256
#define NPIX  65536            // 256*256
#define PCH   128
#define NMODE 32
#define SPEC  1048576          // NMODE*PCH*GRIDN  (one re or im plane set)
#define TWO_PI 6.283185307179586f

__device__ __forceinline__ float gelu_f(float x) {
    // jax.nn.gelu default (tanh approximation)
    float x3 = x * x * x;
    return 0.5f * x * (1.0f + tanhf(0.7978845608028654f * (x + 0.044715f * x3)));
}

__device__ __forceinline__ v8f wmma_f32(v2f a, v2f b, v8f c) {
    return __builtin_amdgcn_wmma_f32_16x16x4_f32(
        /*neg_a=*/false, a, /*neg_b=*/false, b,
        /*c_mod=*/(short)0, c, /*reuse_a=*/false, /*reuse_b=*/false);
}

// ---------------------------------------------------------------------------
// Encoder: v[p, n] = enc_w[p,0]*x0 + enc_w[p,1]*x1 + enc_w[p,2]*u + enc_b[p]
// ---------------------------------------------------------------------------
__global__ void k_encode(const float* __restrict__ u, const float* __restrict__ x,
                         const float* __restrict__ ew, const float* __restrict__ eb,
                         float* __restrict__ v) {
    int n = blockIdx.x * 256 + threadIdx.x;
    float x0 = x[n];
    float x1 = x[NPIX + n];
    float uu = u[n];
    #pragma unroll 4
    for (int p = 0; p < PCH; ++p) {
        v[p * NPIX + n] = ew[p * 3 + 0] * x0 + ew[p * 3 + 1] * x1 + ew[p * 3 + 2] * uu + eb[p];
    }
}

// ---------------------------------------------------------------------------
// Tiled transpose per channel: dst[p, x, y] = src[p, y, x]
// block (32,8), grid (8, 8, 128)
// ---------------------------------------------------------------------------
__global__ void k_transpose(const float* __restrict__ src, float* __restrict__ dst) {
    __shared__ float tile[32][33];
    int p = blockIdx.z;
    int x0 = blockIdx.x * 32, y0 = blockIdx.y * 32;
    #pragma unroll
    for (int q = 0; q < 4; ++q) {
        int yy = threadIdx.y + q * 8;
        tile[yy][threadIdx.x] = src[p * NPIX + (y0 + yy) * GRIDN + x0 + threadIdx.x];
    }
    __syncthreads();
    #pragma unroll
    for (int q = 0; q < 4; ++q) {
        int xx = threadIdx.y + q * 8;
        dst[p * NPIX + (x0 + xx) * GRIDN + y0 + threadIdx.x] = tile[threadIdx.x][xx];
    }
}

// ---------------------------------------------------------------------------
// Truncated rDFT along contiguous rows: for row = p*256 + r,
//   Vh[m] = sum_x src[row, x] * exp(-2*pi*i*m*x/256),  m = 0..31
// dst layout: [re/im][m][p][r]  (im plane at dst + SPEC)
// block 256, grid 32768 (=128*256 rows)
// ---------------------------------------------------------------------------
__global__ void k_dft_rows(const float* __restrict__ src, float* __restrict__ dst) {
    __shared__ float ct[256], st[256], row[256], rr[256], ri[256];
    int t = threadIdx.x;
    float sv, cv;
    sincosf(TWO_PI * (float)t * (1.0f / 256.0f), &sv, &cv);
    ct[t] = cv; st[t] = sv;
    int rowi = blockIdx.x;
    row[t] = src[rowi * GRIDN + t];
    __syncthreads();

    int m = t >> 3, seg = t & 7;
    float ar = 0.0f, ai = 0.0f;
    #pragma unroll 8
    for (int j = 0; j < 32; ++j) {
        int xx = seg * 32 + j;
        int idx = (m * xx) & 255;
        float vv = row[xx];
        ar += vv * ct[idx];
        ai -= vv * st[idx];
    }
    rr[t] = ar; ri[t] = ai;
    __syncthreads();
    if (t < NMODE) {
        float sr = 0.0f, si = 0.0f;
        #pragma unroll
        for (int k = 0; k < 8; ++k) { sr += rr[t * 8 + k]; si += ri[t * 8 + k]; }
        int p = rowi >> 8, r = rowi & 255;
        dst[(t * PCH + p) * GRIDN + r]        = sr;
        dst[SPEC + (t * PCH + p) * GRIDN + r] = si;
    }
}

// ---------------------------------------------------------------------------
// Repack A[l] (complex64 [o][i][m][axis]) -> Apk [axis][m][re/im][o][i]
// grid 16384, block 256
// ---------------------------------------------------------------------------
__global__ void k_repack_a(const float* __restrict__ Al, float* __restrict__ apk) {
    int tid = blockIdx.x * 256 + threadIdx.x;
    int i    = tid & 127;
    int o    = (tid >> 7) & 127;
    int reim = (tid >> 14) & 1;
    int m    = (tid >> 15) & 31;
    int axis = tid >> 20;
    apk[tid] = Al[((((o * PCH + i) * NMODE + m) * 2 + axis) * 2) + reim];
}

// ---------------------------------------------------------------------------
// Per-mode complex channel mix (WMMA f32 16x16x4):
//   Ur = Ar*Vr - Ai*Vi ;  Ui = Ar*Vi + Ai*Vr     ([128x128]x[128x256] per mode)
// block 256 (8 waves = 8 M-tiles), grid (16 n-tiles, 32 modes, 2 axes)
// ---------------------------------------------------------------------------
__global__ void k_modemix(const float* __restrict__ apk,
                          const float* __restrict__ vhx, const float* __restrict__ vhy,
                          float* __restrict__ uhx, float* __restrict__ uhy) {
    int ntile = blockIdx.x, m = blockIdx.y, axis = blockIdx.z;
    int wave = threadIdx.x >> 5, lane = threadIdx.x & 31;
    int lh = lane & 15, hh = lane >> 4;

    const float* Ar = apk + ((axis * NMODE + m) * 2 + 0) * (PCH * PCH);
    const float* Ai = apk + ((axis * NMODE + m) * 2 + 1) * (PCH * PCH);
    const float* vb = axis ? vhy : vhx;
    float* ub       = axis ? uhy : uhx;
    const float* Vr = vb + m * (PCH * GRIDN);
    const float* Vi = vb + SPEC + m * (PCH * GRIDN);

    int rowA = (wave * 16 + lh) * PCH;
    int col  = ntile * 16 + lh;

    v8f cr = {}; v8f ci = {};
    #pragma unroll 4
    for (int k0 = 0; k0 < PCH; k0 += 4) {
        int kk = k0 + 2 * hh;
        v2f a_r = { Ar[rowA + kk], Ar[rowA + kk + 1] };
        v2f a_i = { Ai[rowA + kk], Ai[rowA + kk + 1] };
        v2f a_ni = -a_i;
        v2f b_r = { Vr[kk * GRIDN + col], Vr[(kk + 1) * GRIDN + col] };
        v2f b_i = { Vi[kk * GRIDN + col], Vi[(kk + 1) * GRIDN + col] };
        cr = wmma_f32(a_r,  b_r, cr);
        cr = wmma_f32(a_ni, b_i, cr);
        ci = wmma_f32(a_r,  b_i, ci);
        ci = wmma_f32(a_i,  b_r, ci);
    }
    float* Ur = ub + m * (PCH * GRIDN);
    float* Ui = ub + SPEC + m * (PCH * GRIDN);
    #pragma unroll
    for (int r = 0; r < 8; ++r) {
        int o = wave * 16 + r + 8 * hh;
        Ur[o * GRIDN + col] = cr[r];
        Ui[o * GRIDN + col] = ci[r];
    }
}

// ---------------------------------------------------------------------------
// Fused inverse rDFT for both axes:
//  s[p,y,x] = (1/256)*[ Uhx[0,p,y].re + 2*sum_{m>=1}(Uxr cos(mx) - Uxi sin(mx))
//                     + Uhy[0,p,x].re + 2*sum_{m>=1}(Uyr cos(my) - Uyi sin(my)) ]
// block 256, grid (8 x-tiles, 8 y-tiles, 128 channels)
// ---------------------------------------------------------------------------
__global__ void k_idft_combine(const float* __restrict__ uhx, const float* __restrict__ uhy,
                               float* __restrict__ s) {
    __shared__ float ct[256], st[256];
    __shared__ float uxr[32][32], uxi[32][32], uyr[32][32], uyi[32][32];
    int t = threadIdx.x;
    int p = blockIdx.z, x0 = blockIdx.x * 32, y0 = blockIdx.y * 32;
    float sv, cv;
    sincosf(TWO_PI * (float)t * (1.0f / 256.0f), &sv, &cv);
    ct[t] = cv; st[t] = sv;
    for (int idx = t; idx < 1024; idx += 256) {
        int m = idx >> 5, j = idx & 31;
        uxr[m][j] = uhx[(m * PCH + p) * GRIDN + y0 + j];
        uxi[m][j] = uhx[SPEC + (m * PCH + p) * GRIDN + y0 + j];
        uyr[m][j] = uhy[(m * PCH + p) * GRIDN + x0 + j];
        uyi[m][j] = uhy[SPEC + (m * PCH + p) * GRIDN + x0 + j];
    }
    __syncthreads();

    int tx = t & 31, tyq = t >> 5;
    int x = x0 + tx;
    #pragma unroll
    for (int q = 0; q < 4; ++q) {
        int yl = tyq * 4 + q;
        int y = y0 + yl;
        float accx = uxr[0][yl];
        float accy = uyr[0][tx];
        #pragma unroll 8
        for (int m = 1; m < NMODE; ++m) {
            int ix = (m * x) & 255;
            int iy = (m * y) & 255;
            accx += 2.0f * (uxr[m][yl] * ct[ix] - uxi[m][yl] * st[ix]);
            accy += 2.0f * (uyr[m][tx] * ct[iy] - uyi[m][tx] * st[iy]);
        }
        s[p * NPIX + y * GRIDN + x] = (accx + accy) * (1.0f / 256.0f);
    }
}

// ---------------------------------------------------------------------------
// Pointwise 128x128 GEMM over all pixels (WMMA f32 16x16x4) + bias + GELU
// out[o,n] = (resid? resid[o,n] : 0) + gelu( sum_k W[o,k]*src[k,n] + b[o] )
// block 256 (8 waves = all 8 M-tiles), grid 4096 n-tiles
// ---------------------------------------------------------------------------
__global__ void k_pointwise(const float* __restrict__ W, const float* __restrict__ bias,
                            const float* __restrict__ src, const float* __restrict__ resid,
                            float* __restrict__ out) {
    int ntile = blockIdx.x;
    int wave = threadIdx.x >> 5, lane = threadIdx.x & 31;
    int lh = lane & 15, hh = lane >> 4;
    int rowA = (wave * 16 + lh) * PCH;
    int col = ntile * 16 + lh;

    v8f c = {};
    #pragma unroll 4
    for (int k0 = 0; k0 < PCH; k0 += 4) {
        int kk = k0 + 2 * hh;
        v2f a = { W[rowA + kk], W[rowA + kk + 1] };
        v2f b = { src[kk * NPIX + col], src[(kk + 1) * NPIX + col] };
        c = wmma_f32(a, b, c);
    }
    #pragma unroll
    for (int r = 0; r < 8; ++r) {
        int o = wave * 16 + r + 8 * hh;
        float val = gelu_f(c[r] + bias[o]);
        if (resid) val += resid[o * NPIX + col];
        out[o * NPIX + col] = val;
    }
}

// ---------------------------------------------------------------------------
// Decoder: out[n] = dec_b + sum_p dec_w[p] * v[p,n]
// ---------------------------------------------------------------------------
__global__ void k_decode(const float* __restrict__ v, const float* __restrict__ dw,
                         const float* __restrict__ db, float* __restrict__ out) {
    int n = blockIdx.x * 256 + threadIdx.x;
    float acc = db[0];
    #pragma unroll 4
    for (int p = 0; p < PCH; ++p) acc += dw[p] * v[p * NPIX + n];
    out[n] = acc;
}

// ---------------------------------------------------------------------------
// Host driver
// ---------------------------------------------------------------------------
extern "C" void kernel_launch(void* const* d_in, const int* in_sizes, int n_in,
                              void* d_out, int out_size, void* d_ws, size_t ws_size,
                              hipStream_t stream) {
    const float* u     = (const float*)d_in[0];
    const float* x     = (const float*)d_in[1];
    const float* enc_w = (const float*)d_in[2];
    const float* enc_b = (const float*)d_in[3];
    const float* dec_w = (const float*)d_in[4];
    const float* dec_b = (const float*)d_in[5];
    const float* w1    = (const float*)d_in[6];
    const float* b1    = (const float*)d_in[7];
    const float* w2    = (const float*)d_in[8];
    const float* b2    = (const float*)d_in[9];
    const float* A     = (const float*)d_in[10];   // complex64 interleaved

    float* ws = (float*)d_ws;
    const size_t ACT = (size_t)PCH * NPIX;         // 8,388,608 floats
    float* v   = ws;
    float* t   = ws + ACT;                         // transpose buf, then hidden
    float* s   = ws + 2 * ACT;
    float* vhx = ws + 3 * ACT;                     // 2*SPEC each
    float* vhy = vhx + 2 * SPEC;
    float* uhx = vhy + 2 * SPEC;
    float* uhy = uhx + 2 * SPEC;
    float* apk = uhy + 2 * SPEC;                   // 4,194,304 floats

    (void)in_sizes; (void)n_in; (void)out_size; (void)ws_size;

    k_encode<<<NPIX / 256, 256, 0, stream>>>(u, x, enc_w, enc_b, v);

    const size_t A_LAYER  = (size_t)PCH * PCH * NMODE * 2 * 2;  // floats per layer
    const size_t W_LAYER  = (size_t)PCH * PCH;

    for (int l = 0; l < 4; ++l) {
        // axis-y rows made contiguous via transpose
        k_transpose<<<dim3(8, 8, PCH), dim3(32, 8), 0, stream>>>(v, t);
        // truncated forward DFTs
        k_dft_rows<<<PCH * GRIDN, 256, 0, stream>>>(v, vhx);
        k_dft_rows<<<PCH * GRIDN, 256, 0, stream>>>(t, vhy);
        // repack this layer's spectral weights
        k_repack_a<<<16384, 256, 0, stream>>>(A + (size_t)l * A_LAYER, apk);
        // per-mode complex channel mixing (WMMA)
        k_modemix<<<dim3(16, NMODE, 2), 256, 0, stream>>>(apk, vhx, vhy, uhx, uhy);
        // fused inverse DFTs + axis sum -> s
        k_idft_combine<<<dim3(8, 8, PCH), 256, 0, stream>>>(uhx, uhy, s);
        // h = gelu(W1 s + b1) -> t ;  v = v + gelu(W2 h + b2)
        k_pointwise<<<NPIX / 16, 256, 0, stream>>>(w1 + l * W_LAYER, b1 + l * PCH,
                                                   s, nullptr, t);
        k_pointwise<<<NPIX / 16, 256, 0, stream>>>(w2 + l * W_LAYER, b2 + l * PCH,
                                                   t, v, v);
    }

    k_decode<<<NPIX / 256, 256, 0, stream>>>(v, dec_w, dec_b, (float*)d_out);
}